// Bert_Seq_59055800320845
// MI455X (gfx1250) — compile-verified
//
#include <hip/hip_runtime.h>

typedef __attribute__((ext_vector_type(2))) float v2f;
typedef __attribute__((ext_vector_type(8))) float v8f;

#define S_DIM 512
#define B_DIM 1024
#define T_DIM 64
#define LOG2E 1.4426950408889634f
#define LN2f  0.6931471805599453f
#define ET_STRIDE 68   // 64 + 4 pad -> conflict-free ds_load_b64 across 32 lanes

#if __has_builtin(__builtin_amdgcn_exp2f)
#define FAST_EXP2(x) __builtin_amdgcn_exp2f(x)
#else
#define FAST_EXP2(x) __builtin_exp2f(x)
#endif
#if __has_builtin(__builtin_amdgcn_logf)
#define FAST_LOG2(x) __builtin_amdgcn_logf(x)   // v_log_f32 = log2
#else
#define FAST_LOG2(x) __builtin_log2f(x)
#endif

// ---------------------------------------------------------------------------
// Kernel 1: numerator path score, time-chunked (8 chunks of 64 steps).
// partial[chunk][b] ; chunk 0 also folds in start/end/t=0 terms.
// ---------------------------------------------------------------------------
__global__ void crf_score_kernel(const float* __restrict__ em,
                                 const int* __restrict__ tags,
                                 const int* __restrict__ lengths,
                                 const float* __restrict__ startT,
                                 const float* __restrict__ endT,
                                 const float* __restrict__ trans,
                                 float* __restrict__ partial) {
    int b = blockIdx.x * blockDim.x + threadIdx.x;
    if (b >= B_DIM) return;
    int chunk = blockIdx.y;
    int t0 = chunk * (S_DIM / 8);
    int t1 = t0 + (S_DIM / 8);

    int len = lengths[b];
    float sc = 0.f;
    int prev;
    int tstart = t0;
    if (chunk == 0) {
        int tg0 = tags[b];
        sc = startT[tg0] + em[(size_t)b * T_DIM + tg0];
        int lastTag = tags[(size_t)(len - 1) * B_DIM + b];
        sc += endT[lastTag];
        prev = tg0;
        tstart = 1;
    } else {
        prev = tags[(size_t)(t0 - 1) * B_DIM + b];
    }
    for (int t = tstart; t < t1; ++t) {
        int cur = tags[(size_t)t * B_DIM + b];
        if (t < len) {
            sc += trans[prev * T_DIM + cur] +
                  em[((size_t)t * B_DIM + b) * T_DIM + cur];
        }
        prev = cur;
    }
    partial[(size_t)chunk * B_DIM + b] = sc;
}

// ---------------------------------------------------------------------------
// Kernel 2: forward algorithm via fp32 WMMA. One wave per 16 batches.
//
// State layout (per wave, matches WMMA 16x16 f32 C/D layout):
//   alpha[mt][r] on lane L: batch = b0 + (L%16),
//                           tag   = 16*mt + 8*(L/16) + r
// GEMM per step: D[mt] = sum_kc A(E^T tile mt,kc : 16x4) x B(p^T chunk kc : 4x16)
//   A from LDS (E^T row-major, stride 68), B built from p regs with one
//   xor-16 shuffle pair per chunk.
// ---------------------------------------------------------------------------
__global__ __launch_bounds__(32)
void crf_forward_kernel(const float* __restrict__ em,
                        const int* __restrict__ lengths,
                        const float* __restrict__ startT,
                        const float* __restrict__ endT,
                        const float* __restrict__ trans,
                        float* __restrict__ denom_out) {
    __shared__ float ET[T_DIM * ET_STRIDE];

    const int lane = threadIdx.x;        // 0..31
    const int l16  = lane & 15;
    const int hi   = lane >> 4;          // lane half

    // E^T: ET[j*stride + i] = exp(trans[i][j])
    for (int idx = lane; idx < T_DIM * T_DIM; idx += 32) {
        int i = idx >> 6, j = idx & 63;
        ET[j * ET_STRIDE + i] = FAST_EXP2(trans[idx] * LOG2E);
    }
    __syncthreads();

    const int b   = blockIdx.x * 16 + l16;
    const int len = lengths[b];

    float alpha[4][8];
    // t = 0 : alpha = start + emissions[0]
    {
        const float* eptr = em + (size_t)b * T_DIM;
#pragma unroll
        for (int mt = 0; mt < 4; ++mt) {
            int tbase = mt * 16 + 8 * hi;
            float ev[8];
            *(float4*)&ev[0] = *(const float4*)(eptr + tbase);
            *(float4*)&ev[4] = *(const float4*)(eptr + tbase + 4);
#pragma unroll
            for (int r = 0; r < 8; ++r)
                alpha[mt][r] = startT[tbase + r] + ev[r];
        }
    }

    for (int t = 1; t < S_DIM; ++t) {
        // per-batch max over 64 tags (32 local + partner half)
        float mx = alpha[0][0];
#pragma unroll
        for (int mt = 0; mt < 4; ++mt)
#pragma unroll
            for (int r = 0; r < 8; ++r)
                mx = fmaxf(mx, alpha[mt][r]);
        mx = fmaxf(mx, __shfl_xor(mx, 16, 32));

        // p = exp(alpha - mx)
        float p[4][8];
#pragma unroll
        for (int mt = 0; mt < 4; ++mt)
#pragma unroll
            for (int r = 0; r < 8; ++r)
                p[mt][r] = FAST_EXP2((alpha[mt][r] - mx) * LOG2E);

        // GEMM: acc[mt] = sum_kc A(mt,kc) x B(kc)
        v8f zero = {};
        v8f acc[4];
#pragma unroll
        for (int mt = 0; mt < 4; ++mt) acc[mt] = zero;

#pragma unroll
        for (int kc = 0; kc < 16; ++kc) {
            const int mtk = kc >> 2;
            const int o0  = (kc & 3) * 4;   // tag offset within mt tile
            float v0, v1;
            if (o0 < 8) {
                // low half owns o0,o0+1 ; high half needs partner's o0+2,o0+3
                float s0 = __shfl_xor(p[mtk][o0 + 2], 16, 32);
                float s1 = __shfl_xor(p[mtk][o0 + 3], 16, 32);
                v0 = hi ? s0 : p[mtk][o0];
                v1 = hi ? s1 : p[mtk][o0 + 1];
            } else {
                const int q = o0 - 8;
                // low half needs partner's q,q+1 ; high half owns q+2,q+3
                float s0 = __shfl_xor(p[mtk][q], 16, 32);
                float s1 = __shfl_xor(p[mtk][q + 1], 16, 32);
                v0 = hi ? p[mtk][q + 2] : s0;
                v1 = hi ? p[mtk][q + 3] : s1;
            }
            v2f bfrag;
            bfrag.x = v0;
            bfrag.y = v1;
#pragma unroll
            for (int mt = 0; mt < 4; ++mt) {
                v2f afrag = *(const v2f*)&ET[(mt * 16 + l16) * ET_STRIDE + kc * 4 + 2 * hi];
                acc[mt] = __builtin_amdgcn_wmma_f32_16x16x4_f32(
                    false, afrag, false, bfrag, (short)0, acc[mt], false, false);
            }
        }

        // alpha' = log(acc) + mx + em[t]  (masked by t < len)
        const float* eptr = em + ((size_t)t * B_DIM + b) * T_DIM;
        const bool upd = (t < len);
#pragma unroll
        for (int mt = 0; mt < 4; ++mt) {
            int tbase = mt * 16 + 8 * hi;
            float ev[8];
            *(float4*)&ev[0] = *(const float4*)(eptr + tbase);
            *(float4*)&ev[4] = *(const float4*)(eptr + tbase + 4);
#pragma unroll
            for (int r = 0; r < 8; ++r) {
                float cand = FAST_LOG2(acc[mt][r]) * LN2f + mx + ev[r];
                alpha[mt][r] = upd ? cand : alpha[mt][r];
            }
        }
    }

    // denom = logsumexp(alpha + end)
    float loc = -3.4e38f;
    float fv[4][8];
#pragma unroll
    for (int mt = 0; mt < 4; ++mt) {
        int tbase = mt * 16 + 8 * hi;
#pragma unroll
        for (int r = 0; r < 8; ++r) {
            fv[mt][r] = alpha[mt][r] + endT[tbase + r];
            loc = fmaxf(loc, fv[mt][r]);
        }
    }
    float M = fmaxf(loc, __shfl_xor(loc, 16, 32));
    float s = 0.f;
#pragma unroll
    for (int mt = 0; mt < 4; ++mt)
#pragma unroll
        for (int r = 0; r < 8; ++r)
            s += FAST_EXP2((fv[mt][r] - M) * LOG2E);
    s += __shfl_xor(s, 16, 32);
    float denom = M + FAST_LOG2(s) * LN2f;
    if (hi == 0) denom_out[b] = denom;
}

// ---------------------------------------------------------------------------
// Kernel 3: deterministic scalar reduction  out = sum_b(sum_c part[c][b]-den[b])
// ---------------------------------------------------------------------------
__global__ void crf_reduce_kernel(const float* __restrict__ partial,
                                  const float* __restrict__ denom,
                                  float* __restrict__ out) {
    __shared__ float red[256];
    int tid = threadIdx.x;
    float v = 0.f;
    for (int i = tid; i < B_DIM; i += 256) {
        float sc = 0.f;
        for (int c = 0; c < 8; ++c) sc += partial[(size_t)c * B_DIM + i];
        v += sc - denom[i];
    }
    red[tid] = v;
    __syncthreads();
    for (int sft = 128; sft > 0; sft >>= 1) {
        if (tid < sft) red[tid] += red[tid + sft];
        __syncthreads();
    }
    if (tid == 0) out[0] = red[0];
}

// ---------------------------------------------------------------------------
extern "C" void kernel_launch(void* const* d_in, const int* in_sizes, int n_in,
                              void* d_out, int out_size, void* d_ws, size_t ws_size,
                              hipStream_t stream) {
    const float* em      = (const float*)d_in[0];   // (S,B,T)
    const int*   tags    = (const int*)d_in[1];     // (S,B)
    const int*   lengths = (const int*)d_in[2];     // (B,)
    const float* startT  = (const float*)d_in[3];   // (T,)
    const float* endT    = (const float*)d_in[4];   // (T,)
    const float* trans   = (const float*)d_in[5];   // (T,T)

    float* ws_partial = (float*)d_ws;               // 8 * B floats
    float* ws_denom   = ws_partial + 8 * B_DIM;     // B floats

    // numerator, time-chunked 8x
    dim3 g1(B_DIM / 128, 8);
    crf_score_kernel<<<g1, 128, 0, stream>>>(em, tags, lengths, startT, endT,
                                             trans, ws_partial);
    // forward algorithm: 64 waves, one per 16 batches, spread over WGPs
    crf_forward_kernel<<<B_DIM / 16, 32, 0, stream>>>(em, lengths, startT, endT,
                                                      trans, ws_denom);
    // deterministic final reduce
    crf_reduce_kernel<<<1, 256, 0, stream>>>(ws_partial, ws_denom, (float*)d_out);
}